// DensityLoss_24180665877233
// MI455X (gfx1250) — compile-verified
//
#include <hip/hip_runtime.h>
#include <hip/hip_bf16.h>
#include <stdint.h>

// DensityLoss: out[b,y,x] = loss[b,y,x] * (union-of-boxes ? 10 : 1) / (H*W*B)
// B=8, H=W=1024, N=64 boxes/image. Bandwidth-bound: 64 MB moved -> ~2.8us @ 23.3 TB/s.
// One 256-thread block per (b, y) row. Mask via per-row difference array + block scan
// (O(1) work/pixel). Loss row streamed in with CDNA5 async global->LDS DMA (ASYNCcnt),
// overlapped with the mask computation.

#define WIDTH   1024
#define HEIGHT  1024
#define TPB     256
#define NWAVES  (TPB / 32)
#define BALANCE 10.0f

typedef __attribute__((address_space(3))) float lds_f32_t;

__device__ __forceinline__ unsigned lds_byte_addr(void* p) {
    // generic -> LDS addrspace(3) pointer (32-bit LDS byte offset), then to int
    return (unsigned)(unsigned long long)(lds_f32_t*)p;
}

__global__ __launch_bounds__(TPB) void density_loss_kernel(
    const float* __restrict__ loss,
    const int*   __restrict__ bboxes,
    float*       __restrict__ out,
    int n_boxes, float scale_bg, float scale_fg)
{
    __shared__ float lds_loss[WIDTH];
    __shared__ int   diff[WIDTH + 1];
    __shared__ int   wave_sums[NWAVES];

    const int tid  = threadIdx.x;          // 0..255
    const int lane = tid & 31;             // wave32
    const int wid  = tid >> 5;
    const int y    = blockIdx.x;           // row
    const int b    = blockIdx.y;           // image
    const int base = tid * 4;              // 4 pixels / thread

    // ---- 1) Kick off async DMA: 16B of this loss row -> LDS (ASYNCcnt path) ----
    const size_t row_off = ((size_t)b * HEIGHT + (size_t)y) * WIDTH;
    unsigned long long gaddr = (unsigned long long)(const void*)(loss + row_off + base);
    unsigned laddr = lds_byte_addr(&lds_loss[base]);
    asm volatile("global_load_async_to_lds_b128 %0, %1, off"
                 :: "v"(laddr), "v"(gaddr) : "memory");

    // ---- 2) Zero the difference array (overlapped with DMA) ----
    *(int4*)&diff[base] = make_int4(0, 0, 0, 0);
    if (tid == 0) diff[WIDTH] = 0;
    __syncthreads();

    // ---- 3) Scatter +-1 x-interval deltas for boxes covering row y ----
    const int* bb = bboxes + (size_t)b * (size_t)n_boxes * 4;
    for (int i = tid; i < n_boxes; i += TPB) {
        int x1 = bb[i * 4 + 0], y1 = bb[i * 4 + 1];
        int x2 = bb[i * 4 + 2], y2 = bb[i * 4 + 3];
        x1 = min(max(x1, 0), WIDTH);   x2 = min(max(x2, 0), WIDTH);
        y1 = min(max(y1, 0), HEIGHT);  y2 = min(max(y2, 0), HEIGHT);
        if (y1 <= y && y < y2) {       // box active at this row
            atomicAdd(&diff[x1],  1);
            atomicAdd(&diff[x2], -1);
        }
    }
    __syncthreads();

    // ---- 4) Block-wide inclusive scan of diff -> per-pixel coverage count ----
    int d0 = diff[base], d1 = diff[base + 1], d2 = diff[base + 2], d3 = diff[base + 3];
    int s0 = d0, s1 = s0 + d1, s2 = s1 + d2, s3 = s2 + d3;

    int v = s3;                                   // wave32 inclusive scan of thread sums
    #pragma unroll
    for (int off = 1; off < 32; off <<= 1) {
        int n = __shfl_up(v, off, 32);
        if (lane >= off) v += n;
    }
    if (lane == 31) wave_sums[wid] = v;
    __syncthreads();

    int woff = 0;                                 // sum of preceding waves' totals
    #pragma unroll
    for (int wI = 0; wI < NWAVES; ++wI) woff += (wI < wid) ? wave_sums[wI] : 0;
    const int excl = woff + (v - s3);             // exclusive prefix for this thread

    // ---- 5) Wait for DMA, apply weights (normalization folded in), store ----
    asm volatile("s_wait_asynccnt 0" ::: "memory");
    float4 lv = *(const float4*)&lds_loss[base];  // ds_load_b128 of our own DMA'd bytes

    float w0 = ((excl + s0) > 0) ? scale_fg : scale_bg;
    float w1 = ((excl + s1) > 0) ? scale_fg : scale_bg;
    float w2 = ((excl + s2) > 0) ? scale_fg : scale_bg;
    float w3 = ((excl + s3) > 0) ? scale_fg : scale_bg;

    float4 o;
    o.x = lv.x * w0;
    o.y = lv.y * w1;
    o.z = lv.z * w2;
    o.w = lv.w * w3;
    *(float4*)(out + row_off + base) = o;         // global_store_b128
}

extern "C" void kernel_launch(void* const* d_in, const int* in_sizes, int n_in,
                              void* d_out, int out_size, void* d_ws, size_t ws_size,
                              hipStream_t stream) {
    const float* loss   = (const float*)d_in[0];
    // d_in[1] (pred_densities) is dead in the reference output — never read.
    const int*   bboxes = (const int*)d_in[2];
    float*       out    = (float*)d_out;

    const int B = in_sizes[0] / (HEIGHT * WIDTH);            // 8
    const int N = in_sizes[2] / (B * 4);                     // 64
    const float scale = 1.0f / ((float)HEIGHT * (float)WIDTH * (float)B);

    dim3 grid(HEIGHT, B);
    density_loss_kernel<<<grid, TPB, 0, stream>>>(loss, bboxes, out,
                                                  N, scale, BALANCE * scale);
}